// GreedyFeatureInit_47244640256005
// MI455X (gfx1250) — compile-verified
//
#include <hip/hip_runtime.h>
#include <hip/hip_bf16.h>
#include <math.h>

// Problem constants (fixed by the reference's setup_inputs)
#define BB 16
#define NN 8192
#define DD 768
#define NSLOTS 24
#define EPS 1e-12f

typedef float v2f __attribute__((ext_vector_type(2)));
typedef float v8f __attribute__((ext_vector_type(8)));

// ---------------------------------------------------------------------------
// Kernel 0: per-row L2 norms of features, and mask init to 1.0
// One wave (32 lanes) per row; coalesced loads; __shfl_xor wave reduction.
// grid = (B*N/8) blocks of 256 threads (8 waves/block).
// ---------------------------------------------------------------------------
__global__ void norms_init_kernel(const float* __restrict__ features,
                                  float* __restrict__ norms,
                                  float* __restrict__ mask) {
  int wave = threadIdx.x >> 5;
  int lane = threadIdx.x & 31;
  size_t row = (size_t)blockIdx.x * 8 + wave;   // 0 .. B*N-1
  const float* f = features + row * DD;
  float ss = 0.0f;
  #pragma unroll 6
  for (int i = lane; i < DD; i += 32) {
    float v = f[i];
    ss += v * v;
  }
  #pragma unroll
  for (int off = 16; off > 0; off >>= 1)
    ss += __shfl_xor(ss, off, 32);
  if (lane == 0) {
    norms[row] = sqrtf(ss);
    mask[row]  = 1.0f;
  }
}

// ---------------------------------------------------------------------------
// Kernel 1 (per slot): masked argmax over N per batch, then normalize the
// selected feature row. One block per batch, 256 threads.
// Tie-break = lowest index (matches jnp.argmax first-occurrence).
// Writes: sel_norm[b, 0:768] (unit vector), d_out[b, s, 0:768] (raw row).
// ---------------------------------------------------------------------------
__global__ void argmax_select_kernel(const float* __restrict__ features,
                                     const float* __restrict__ norms,
                                     const float* __restrict__ mask,
                                     float* __restrict__ sel_norm,
                                     float* __restrict__ out,
                                     int s) {
  __shared__ float s_val[256];
  __shared__ int   s_idx[256];
  const int b   = blockIdx.x;
  const int tid = threadIdx.x;

  // Phase A: strided scan + LDS reduce for (max val, first idx)
  float best = -1.0f;
  int   bidx = 0;
  for (int n = tid; n < NN; n += 256) {
    float v = norms[(size_t)b * NN + n] * mask[(size_t)b * NN + n];
    if (v > best) { best = v; bidx = n; }   // strict > keeps earliest in-thread
  }
  s_val[tid] = best;
  s_idx[tid] = bidx;
  __syncthreads();
  #pragma unroll
  for (int off = 128; off > 0; off >>= 1) {
    if (tid < off) {
      float v2 = s_val[tid + off];
      int   i2 = s_idx[tid + off];
      if (v2 > s_val[tid] || (v2 == s_val[tid] && i2 < s_idx[tid])) {
        s_val[tid] = v2;
        s_idx[tid] = i2;
      }
    }
    __syncthreads();
  }
  const int idx = s_idx[0];
  __syncthreads();   // protect s_val before reuse

  // Phase B: load selected row (3 elems/thread), sum-of-squares reduce
  const float* sel = features + ((size_t)b * NN + idx) * (size_t)DD;
  float vals[3];
  float ss = 0.0f;
  #pragma unroll
  for (int i = 0; i < 3; ++i) {
    vals[i] = sel[tid + 256 * i];
    ss += vals[i] * vals[i];
  }
  s_val[tid] = ss;
  __syncthreads();
  #pragma unroll
  for (int off = 128; off > 0; off >>= 1) {
    if (tid < off) s_val[tid] += s_val[tid + off];
    __syncthreads();
  }
  const float inv = 1.0f / fmaxf(sqrtf(s_val[0]), EPS);
  #pragma unroll
  for (int i = 0; i < 3; ++i) {
    int d = tid + 256 * i;
    sel_norm[(size_t)b * DD + d] = vals[i] * inv;
    out[((size_t)b * NSLOTS + s) * DD + d] = vals[i];   // aggregate=False: raw row
  }
}

// ---------------------------------------------------------------------------
// Kernel 2 (per slot, slots 0..22): similarity + mask update via
// V_WMMA_F32_16X16X4_F32.  One wave handles a 16-row tile of features.
//   A fragment (16x4 f32): lanes 0-15 hold M=0..15 with VGPR0/1 = K0/K1,
//                          lanes 16-31 hold K2/K3.
//   B fragment (4x16 f32): sel_norm chunk broadcast into every column, so
//                          every column of C accumulates the same dots.
//   C fragment: VGPR v, lanes 0-15 -> M=v ; lanes 16-31 -> M=8+v.
// grid = (N/128, B), block = 256 (8 waves -> 128 rows per block).
// ---------------------------------------------------------------------------
__global__ void sim_mask_kernel(const float* __restrict__ features,
                                const float* __restrict__ norms,
                                const float* __restrict__ sel_norm,
                                float* __restrict__ mask) {
  __shared__ float s_sel[DD];
  __shared__ float s_sim[8][16];

  const int b = blockIdx.y;

  // Stage the selected unit vector for this batch in LDS (broadcast source)
  for (int i = threadIdx.x; i < DD; i += 256)
    s_sel[i] = sel_norm[(size_t)b * DD + i];
  __syncthreads();

  const int wave = threadIdx.x >> 5;   // 0..7
  const int lane = threadIdx.x & 31;
  const int hi   = lane >> 4;          // 0 or 1
  const int lm   = lane & 15;
  const int row0 = blockIdx.x * 128 + wave * 16;

  const float* fbase = features + ((size_t)b * NN + row0 + lm) * (size_t)DD;

  v8f c = {0.f, 0.f, 0.f, 0.f, 0.f, 0.f, 0.f, 0.f};
  #pragma unroll 8
  for (int k0 = 0; k0 < DD; k0 += 4) {
    const int col = k0 + 2 * hi;                    // this lane's K pair
    v2f a  = *(const v2f*)(fbase + col);            // A[M=lm][K pair]
    v2f bv = *(const v2f*)(&s_sel[col]);            // B[K pair][all N cols]
    c = __builtin_amdgcn_wmma_f32_16x16x4_f32(
        /*neg_a=*/false, a, /*neg_b=*/false, bv,
        /*c_mod=*/(short)0, c, /*reuse_a=*/false, /*reuse_b=*/false);
  }

  // Every column of C is identical; lane 0 holds rows 0..7, lane 16 rows 8..15
  if (lm == 0) {
    #pragma unroll
    for (int v = 0; v < 8; ++v) s_sim[wave][hi * 8 + v] = c[v];
  }
  __syncthreads();

  if (lane < 16) {
    const int n = row0 + lane;
    const size_t mi = (size_t)b * NN + n;
    float sim  = s_sim[wave][lane] / fmaxf(norms[mi], EPS);
    float simc = fminf(fmaxf(sim, 0.0f), 1.0f);
    mask[mi] = mask[mi] * (1.0f - simc);
  }
}

// ---------------------------------------------------------------------------
extern "C" void kernel_launch(void* const* d_in, const int* in_sizes, int n_in,
                              void* d_out, int out_size, void* d_ws, size_t ws_size,
                              hipStream_t stream) {
  (void)in_sizes; (void)n_in; (void)out_size; (void)ws_size;
  // inputs: d_in[0] = batch_size (scalar int, fixed 16), d_in[1] = features f32 [16,8192,768]
  const float* features = (const float*)d_in[1];
  float* out = (float*)d_out;

  // workspace layout
  char* ws = (char*)d_ws;
  float* norms    = (float*)(ws);                                   // B*N
  float* mask     = (float*)(ws + (size_t)BB * NN * sizeof(float)); // B*N
  float* sel_norm = (float*)(ws + (size_t)2 * BB * NN * sizeof(float)); // B*D

  // 1) norms + mask init: one wave per row, 8 waves/block
  norms_init_kernel<<<(BB * NN) / 8, 256, 0, stream>>>(features, norms, mask);

  // 2) 24 sequential greedy rounds
  for (int s = 0; s < NSLOTS; ++s) {
    argmax_select_kernel<<<BB, 256, 0, stream>>>(features, norms, mask,
                                                 sel_norm, out, s);
    if (s < NSLOTS - 1) {   // last round's mask update is dead in the reference
      sim_mask_kernel<<<dim3(NN / 128, BB), 256, 0, stream>>>(features, norms,
                                                              sel_norm, mask);
    }
  }
}